// graph_56006373539875
// MI455X (gfx1250) — compile-verified
//
#include <hip/hip_runtime.h>

// Reference: out = external_forces; for each edge e:
//   v = points[edge_b[e]] - points[edge_a[e]]
//   fo = -force[e] * v / |v|
//   out[edge_a[e]] += fo ; out[edge_b[e]] -= fo
//
// Bottleneck analysis (MI455X): 6.4M edges -> 77 MB streaming reads (~3.3 us
// at 23.3 TB/s) but 38.4M f32 atomic adds, which dominate. points (1.2 MB)
// and out (1.2 MB) stay resident in the 192 MB L2; the edge stream is read
// non-temporally so it cannot evict them. Atomics are relaxed/agent/no-return
// -> native global_atomic_add_f32 (verified: no CAS loop in round-1 histogram).

// ---------------------------------------------------------------------------
// Edge kernel FIRST in the module so the disasm snippet shows it.
// ---------------------------------------------------------------------------
__global__ void __launch_bounds__(256)
graph_edge_scatter_kernel(const float* __restrict__ points,
                          const float* __restrict__ force,
                          const int*   __restrict__ edge_a,
                          const int*   __restrict__ edge_b,
                          float* __restrict__ out,
                          int n_edges)
{
    const int stride = gridDim.x * blockDim.x;
    for (int i = blockIdx.x * blockDim.x + threadIdx.x; i < n_edges; i += stride) {
        // Branchless speculative prefetch of the next grid-stride tile
        // (global_prefetch_b8). Clamp instead of branching: prefetch is
        // side-effect free, so re-touching a valid address is harmless.
        int pf = i + stride;
        pf = (pf < n_edges) ? pf : i;
        __builtin_prefetch(&edge_a[pf], 0, 0);
        __builtin_prefetch(&edge_b[pf], 0, 0);
        __builtin_prefetch(&force[pf],  0, 0);

        // Non-temporal streaming reads (one-touch data; keep L2 for points/out).
        int   a = __builtin_nontemporal_load(&edge_a[i]);
        int   b = __builtin_nontemporal_load(&edge_b[i]);
        float f = __builtin_nontemporal_load(&force[i]);

        // Gathers from the L2-resident points array (regular temporal loads).
        const float* pa = points + 3 * a;
        const float* pb = points + 3 * b;
        float ax = pa[0], ay = pa[1], az = pa[2];
        float bx = pb[0], by = pb[1], bz = pb[2];

        float vx = bx - ax;
        float vy = by - ay;
        float vz = bz - az;

        // Accurate normalize (VALU idle relative to the atomic bound).
        float len = sqrtf(fmaf(vx, vx, fmaf(vy, vy, vz * vz)));
        float s   = -f / len;

        float fx = s * vx;
        float fy = s * vy;
        float fz = s * vz;

        float* oa = out + 3 * a;
        float* ob = out + 3 * b;
        (void)__hip_atomic_fetch_add(&oa[0],  fx, __ATOMIC_RELAXED, __HIP_MEMORY_SCOPE_AGENT);
        (void)__hip_atomic_fetch_add(&oa[1],  fy, __ATOMIC_RELAXED, __HIP_MEMORY_SCOPE_AGENT);
        (void)__hip_atomic_fetch_add(&oa[2],  fz, __ATOMIC_RELAXED, __HIP_MEMORY_SCOPE_AGENT);
        (void)__hip_atomic_fetch_add(&ob[0], -fx, __ATOMIC_RELAXED, __HIP_MEMORY_SCOPE_AGENT);
        (void)__hip_atomic_fetch_add(&ob[1], -fy, __ATOMIC_RELAXED, __HIP_MEMORY_SCOPE_AGENT);
        (void)__hip_atomic_fetch_add(&ob[2], -fz, __ATOMIC_RELAXED, __HIP_MEMORY_SCOPE_AGENT);
    }
}

// ---------------------------------------------------------------------------
// out = external_forces. Vec4 fast path (n % 4 == 0, the actual case) and a
// scalar fallback kernel; the host picks. No in-kernel tail loop -> no
// scalarized branchy mess in the module.
// ---------------------------------------------------------------------------
__global__ void __launch_bounds__(256)
graph_init_out_vec4_kernel(const float4* __restrict__ ext,
                           float4* __restrict__ out,
                           int n4)
{
    int i = blockIdx.x * blockDim.x + threadIdx.x;
    if (i < n4) out[i] = ext[i];
}

__global__ void __launch_bounds__(256)
graph_init_out_scalar_kernel(const float* __restrict__ ext,
                             float* __restrict__ out,
                             int n)
{
    int i = blockIdx.x * blockDim.x + threadIdx.x;
    if (i < n) out[i] = ext[i];
}

// ---------------------------------------------------------------------------
// Inputs (setup_inputs order):
//   d_in[0] points           [100000*3] f32
//   d_in[1] external_forces  [100000*3] f32
//   d_in[2] force            [6400000]  f32
//   d_in[3] edge_a           [6400000]  int
//   d_in[4] edge_b           [6400000]  int
// d_out: [100000*3] f32
// ---------------------------------------------------------------------------
extern "C" void kernel_launch(void* const* d_in, const int* in_sizes, int n_in,
                              void* d_out, int out_size, void* d_ws, size_t ws_size,
                              hipStream_t stream)
{
    const float* points = (const float*)d_in[0];
    const float* ext    = (const float*)d_in[1];
    const float* force  = (const float*)d_in[2];
    const int*   edge_a = (const int*)d_in[3];
    const int*   edge_b = (const int*)d_in[4];
    float*       out    = (float*)d_out;

    const int n_out   = out_size;     // 300000
    const int n_edges = in_sizes[2];  // 6400000

    // Kernel 1: out = external_forces.
    if ((n_out & 3) == 0) {
        int n4     = n_out >> 2;                  // 75000 float4s
        int blocks = (n4 + 255) / 256;
        graph_init_out_vec4_kernel<<<blocks, 256, 0, stream>>>(
            (const float4*)ext, (float4*)out, n4);
    } else {
        int blocks = (n_out + 255) / 256;
        graph_init_out_scalar_kernel<<<blocks, 256, 0, stream>>>(ext, out, n_out);
    }

    // Kernel 2: grid-stride over edges (~8 edges/thread) so the
    // one-grid-stride prefetch distance is meaningful.
    {
        const int EPT    = 8;
        int blocks = (n_edges + 256 * EPT - 1) / (256 * EPT);  // 3125 blocks
        graph_edge_scatter_kernel<<<blocks, 256, 0, stream>>>(
            points, force, edge_a, edge_b, out, n_edges);
    }
}